// UnifiedMorphoLayerExtSE_83193516523623
// MI455X (gfx1250) — compile-verified
//
#include <hip/hip_runtime.h>
#include <stdint.h>

// Morphological dilation (max-plus 5x5, stride 1, pad 2) for
// x:[16,64,128,128], weight:[16,64,5,5], bias:[16,64], sign:[16].
// y[b,c,i,j] = (max_{p,q} fma(x[b,c,i+p-2,j+q-2], s_b, w[b,c,p,q]) + bias) * s_b
// with s_b = sign[b] if |sign[b]|>=1e-7 else 1, and zero halo (0*s+w == w,
// exactly matching the reference's scale-then-zero-pad).

namespace {
constexpr int Bb   = 16;
constexpr int Cc   = 64;
constexpr int Hh   = 128;
constexpr int Ww   = 128;
constexpr int TILE = 32;   // 32x32 outputs per block
constexpr int INDM = 36;   // TILE + 2*pad
constexpr int LSTR = 40;   // padded LDS row stride (floats) -> conflict-free
}

__device__ __forceinline__ float max5(float a, float b, float c, float d, float e) {
    // fmax(fmax(x,y),z) pattern -> v_max3 ; two max3 per 5-way reduce
    float m = fmaxf(fmaxf(a, b), c);
    return fmaxf(fmaxf(m, d), e);
}

__global__ void __launch_bounds__(256)
morpho5x5_kernel(const float* __restrict__ x,
                 const float* __restrict__ weight,
                 const float* __restrict__ bias,
                 const float* __restrict__ sign,
                 float* __restrict__ out)
{
    __shared__ float tile[INDM * LSTR];  // 5760 B raw-x tile with zero halo

    const int tid = threadIdx.x;
    const int bc  = blockIdx.z;              // fused (b,c) plane, 0..1023 (uniform)
    const int ox0 = blockIdx.x * TILE;
    const int oy0 = blockIdx.y * TILE;

    // Per-plane parameters: bc is wave-uniform, so these become SGPR s_loads
    // (KMcnt) issued up front and overlapped with the async tile staging.
    const float* wp = weight + bc * 25;
    float w[25];
    #pragma unroll
    for (int k = 0; k < 25; ++k) w[k] = wp[k];
    const float bi = bias[bc];
    const float sg = sign[bc >> 6];          // b = bc / C
    const float s  = (fabsf(sg) >= 1e-7f) ? sg : 1.0f;

    // Stage 36x36 input tile: async global->LDS for in-range elements
    // (ASYNCcnt path, no VGPR round trip), ds_store zeros for the halo.
    const float* xp = x + (size_t)bc * (Hh * Ww);
    #pragma unroll
    for (int k = 0; k < 6; ++k) {
        int idx = tid + k * 256;
        if (idx < INDM * INDM) {
            int r  = idx / INDM;
            int c  = idx - r * INDM;
            int gy = oy0 - 2 + r;
            int gx = ox0 - 2 + c;
            uint32_t laddr = (uint32_t)(uintptr_t)&tile[r * LSTR + c];
            if ((unsigned)gy < (unsigned)Hh && (unsigned)gx < (unsigned)Ww) {
                const float* gp = xp + (gy * Ww + gx);
                asm volatile("global_load_async_to_lds_b32 %0, %1, off"
                             :: "v"(laddr), "v"(gp) : "memory");
            } else {
                tile[r * LSTR + c] = 0.0f;
            }
        }
    }
    asm volatile("s_wait_asynccnt 0x0" ::: "memory");  // my wave's async copies done
    __syncthreads();                                   // everyone's copies visible

    // 256 threads = 32 (x) x 8 (y); each thread: 4 consecutive output rows,
    // one column -> 8x5 input window feeds 4*25 taps (10 LDS loads/output).
    const int tx = tid & 31;
    const int ty = tid >> 5;
    const int r0 = ty * 4;                             // top tile-row of window

    float v[8][5];
    #pragma unroll
    for (int r = 0; r < 8; ++r)
        #pragma unroll
        for (int q = 0; q < 5; ++q)
            v[r][q] = tile[(r0 + r) * LSTR + tx + q];  // stride-1 lanes: no bank conflicts

    float* op = out + (size_t)bc * (Hh * Ww) + (size_t)(oy0 + r0) * Ww + (ox0 + tx);
    #pragma unroll
    for (int rr = 0; rr < 4; ++rr) {
        float pr[5];
        #pragma unroll
        for (int p = 0; p < 5; ++p) {
            // v_fma_f32 with scalar s and scalar w operands (SGPR const-bus <= 2)
            float t0 = fmaf(v[rr + p][0], s, w[p * 5 + 0]);
            float t1 = fmaf(v[rr + p][1], s, w[p * 5 + 1]);
            float t2 = fmaf(v[rr + p][2], s, w[p * 5 + 2]);
            float t3 = fmaf(v[rr + p][3], s, w[p * 5 + 3]);
            float t4 = fmaf(v[rr + p][4], s, w[p * 5 + 4]);
            pr[p] = max5(t0, t1, t2, t3, t4);
        }
        float m = max5(pr[0], pr[1], pr[2], pr[3], pr[4]);
        op[(size_t)rr * Ww] = (m + bi) * s;            // coalesced b32 stores
    }
}

extern "C" void kernel_launch(void* const* d_in, const int* in_sizes, int n_in,
                              void* d_out, int out_size, void* d_ws, size_t ws_size,
                              hipStream_t stream) {
    const float* x      = (const float*)d_in[0];
    const float* weight = (const float*)d_in[1];
    const float* bias   = (const float*)d_in[2];
    const float* sign   = (const float*)d_in[3];
    // d_in[4]=padding(2), d_in[5]=stride(1): compile-time constants here.
    float* out = (float*)d_out;

    dim3 grid(Ww / TILE, Hh / TILE, Bb * Cc);  // (4, 4, 1024)
    morpho5x5_kernel<<<grid, dim3(256), 0, stream>>>(x, weight, bias, sign, out);
}